// SpatialAttentionLayer_23381801959766
// MI455X (gfx1250) — compile-verified
//
#include <hip/hip_runtime.h>
#include <math.h>

typedef __attribute__((ext_vector_type(2))) float v2f;
typedef __attribute__((ext_vector_type(4))) float f4;
typedef __attribute__((ext_vector_type(8))) float v8f;

#define S_SENSORS 272
#define J_OUT     128
#define H_HARM    32
#define HALF_K    1024          /* H*H; total K = 2048 (cos half + sin half) */
#define TWO_PI_F  6.28318530717958647692f

// ---------------------------------------------------------------------------
// Kernel 1: logits GEMM via V_WMMA_F32_16X16X4_F32, softmax row-sum -> alpha[s]
// grid = 17 blocks (M tiles of 16 sensors), 256 threads = 8 waves (N tiles of 16 ch)
// ---------------------------------------------------------------------------
__global__ __launch_bounds__(256) void attn_alpha_kernel(
    const float* __restrict__ locs,   // [S,2]
    const float* __restrict__ fw,     // [J,H,H,2]  (.x = Wr/cos, .y = Wi/sin)
    float* __restrict__ alpha)        // [S]
{
    __shared__ float s_cx[16][33];    // cos(2*pi*k*x_s)
    __shared__ float s_sx[16][33];    // sin(2*pi*k*x_s)
    __shared__ float s_cy[16][33];    // cos(2*pi*l*y_s)
    __shared__ float s_sy[16][33];    // sin(2*pi*l*y_s)
    __shared__ float s_attn[16][J_OUT + 1];

    const int tid   = threadIdx.x;
    const int lane  = tid & 31;
    const int halfl = lane & 15;
    const int wave  = tid >> 5;       // N tile (output-channel tile)
    const int mt    = blockIdx.x;     // M tile (sensor tile)

    // --- normalize sensor locations: global min/max (272 elems, trivially cheap) ---
    float lox = locs[0], hix = locs[0], loy = locs[1], hiy = locs[1];
    for (int s = 1; s < S_SENSORS; ++s) {
        float x = locs[2 * s], y = locs[2 * s + 1];
        lox = fminf(lox, x); hix = fmaxf(hix, x);
        loy = fminf(loy, y); hiy = fmaxf(hiy, y);
    }
    const float invx = 1.0f / (hix - lox);
    const float invy = 1.0f / (hiy - loy);

    // --- per-sensor harmonic tables for this block's 16 sensors ---
    for (int idx = tid; idx < 16 * H_HARM; idx += 256) {
        int ss = idx >> 5, h = idx & 31;
        int s  = mt * 16 + ss;
        float x = (locs[2 * s]     - lox) * invx;
        float y = (locs[2 * s + 1] - loy) * invy;
        float ax = TWO_PI_F * x * (float)h;
        float ay = TWO_PI_F * y * (float)h;
        s_cx[ss][h] = cosf(ax);  s_sx[ss][h] = sinf(ax);
        s_cy[ss][h] = cosf(ay);  s_sy[ss][h] = sinf(ay);
    }
    __syncthreads();

    // --- WMMA GEMM: D += A_cos*B_cos + A_sin*B_sin over K, 4 at a time ---
    // A 16x4 f32 layout: lanes 0-15 -> M, v0/v1 = K 0/1; lanes 16-31 -> K 2/3.
    // B 4x16 f32 layout: lanes 0-15 -> N, v0/v1 = K 0/1; lanes 16-31 -> K 2/3.
    const int    jcol = wave * 16 + halfl;                   // output channel
    const float2* __restrict__ wrow =
        (const float2*)(fw + (size_t)jcol * (size_t)(2 * HALF_K));
    const int koff = (lane >= 16) ? 2 : 0;

    v8f acc = {};
    for (int kk = 0; kk < HALF_K; kk += 4) {
        v2f ac, as, bc, bs;
#pragma unroll
        for (int e = 0; e < 2; ++e) {
            int   cc = kk + koff + e;
            int   k  = cc >> 5, l = cc & 31;
            float cA = s_cx[halfl][k], sA = s_sx[halfl][k];
            float cB = s_cy[halfl][l], sB = s_sy[halfl][l];
            ac[e] = cA * cB - sA * sB;     // cos(2*pi*(k*x + l*y))
            as[e] = sA * cB + cA * sB;     // sin(2*pi*(k*x + l*y))
            float2 w = wrow[cc];           // {Wr, Wi} for (k,l)
            bc[e] = w.x;
            bs[e] = w.y;
        }
        acc = __builtin_amdgcn_wmma_f32_16x16x4_f32(
                  false, ac, false, bc, (short)0, acc, false, false);
        acc = __builtin_amdgcn_wmma_f32_16x16x4_f32(
                  false, as, false, bs, (short)0, acc, false, false);
    }

    // --- C fragment -> LDS (16x16 f32 C/D layout: VGPR v -> M=v / v+8) ---
#pragma unroll
    for (int v = 0; v < 8; ++v) {
        int row = (lane < 16) ? v : (v + 8);
        s_attn[row][wave * 16 + halfl] = acc[v];
    }
    __syncthreads();

    // --- softmax over J then sum (== alpha_s, ~1.0), matching reference order ---
    if (tid < 16) {
        float m = s_attn[tid][0];
        for (int j = 1; j < J_OUT; ++j) m = fmaxf(m, s_attn[tid][j]);
        float d = 0.0f;
        for (int j = 0; j < J_OUT; ++j) d += __expf(s_attn[tid][j] - m);
        float asum = 0.0f;
        for (int j = 0; j < J_OUT; ++j) asum += __expf(s_attn[tid][j] - m) / d;
        alpha[mt * 16 + tid] = asum;
    }
}

// ---------------------------------------------------------------------------
// Kernel 2: out[s,b,t] = alpha[s] * X[s,b,t].  Pure HBM-streaming copy, B128
// nontemporal loads/stores. Each block covers 1024 contiguous elements, and
// B*T = 131072 is a multiple of 1024, so the whole block shares one sensor.
// ---------------------------------------------------------------------------
__global__ __launch_bounds__(256) void scale_copy_kernel(
    const f4* __restrict__ X4,
    const float* __restrict__ alpha,
    f4* __restrict__ out4)
{
    const size_t i4 = (size_t)blockIdx.x * 256 + threadIdx.x;
    const float  a  = alpha[blockIdx.x >> 7];   // 131072/1024 = 128 blocks/sensor
    f4 v = __builtin_nontemporal_load(&X4[i4]);
    v *= a;
    __builtin_nontemporal_store(v, &out4[i4]);
}

extern "C" void kernel_launch(void* const* d_in, const int* in_sizes, int n_in,
                              void* d_out, int out_size, void* d_ws, size_t ws_size,
                              hipStream_t stream) {
    const float* X    = (const float*)d_in[0];  // [272,128,1024]
    const float* locs = (const float*)d_in[1];  // [272,2]
    const float* fw   = (const float*)d_in[2];  // [128,32,32,2]
    float* alpha = (float*)d_ws;                // 272 floats of scratch

    attn_alpha_kernel<<<S_SENSORS / 16, 256, 0, stream>>>(locs, fw, alpha);

    const int n4     = out_size / 4;            // 8,912,896 float4s
    const int blocks = n4 / 256;                // 34,816 blocks, exact
    scale_copy_kernel<<<blocks, 256, 0, stream>>>((const f4*)X, alpha, (f4*)d_out);
}